// NegativeSampleLoss_77000173683133
// MI455X (gfx1250) — compile-verified
//
#include <hip/hip_runtime.h>
#include <hip/hip_bf16.h>

// Problem constants (from reference): B=2048, T=50, N=100, V=50000, D=512
#define BB 2048
#define TT 50
#define NN 100
#define DD 512
#define ROWS 150          // T + N real rows per example
#define NWAVES 10         // one wave per 16-row tile (160 padded rows)
#define BLOCK (NWAVES * 32)

typedef __attribute__((ext_vector_type(2))) float v2f;
typedef __attribute__((ext_vector_type(8))) float v8f;

__device__ __forceinline__ float log_sigmoid(float x) {
    // stable: ls(x) = min(x,0) - log1p(exp(-|x|))
    return fminf(x, 0.0f) - log1pf(expf(-fabsf(x)));
}

__global__ __launch_bounds__(BLOCK)
void nsl_main_kernel(const float* __restrict__ feat,
                     const int* __restrict__ targets,
                     const int* __restrict__ noises,
                     const float* __restrict__ W,
                     float* __restrict__ partial) {
    __shared__ float fsh[DD];
    __shared__ float wsum[NWAVES];

    const int b    = blockIdx.x;
    const int tid  = threadIdx.x;
    const int wave = tid >> 5;
    const int lane = tid & 31;

    // ---- Stage feature vector into LDS via async global->LDS DMA path ----
    // One b64 per thread for tid<256 covers 512 floats. LDS operand is the
    // wave-relative byte offset = low 32 bits of the generic LDS pointer.
    if (tid < DD / 2) {
        unsigned lds_off      = (unsigned)(size_t)(&fsh[2 * tid]);
        const float* gsrc     = feat + (size_t)b * DD + 2 * tid;
        asm volatile("global_load_async_to_lds_b64 %0, %1, off"
                     :: "v"(lds_off), "v"(gsrc)
                     : "memory");
    }
    asm volatile("s_wait_asynccnt 0x0" ::: "memory");
    __syncthreads();

    // This wave's tile: rows [16*wave, 16*wave+15] of the padded 160-row space.
    // WMMA orientation: C = A(16x4) x B(4x16).
    //   B[k][n] = W[rowIdx(n)][kbase+k]  (columns of B = gathered W rows)
    //   A[0][k] = fsh[kbase+k], A rows 1..15 = 0
    // => C[0][n] accumulates dot(W[rowIdx(n)], feature) across kbase chunks.
    //
    // B VGPR layout (32-bit, 4x16): lanes 0-15 hold K={0,1}, lanes 16-31 hold
    // K={2,3}. So each lane loads a float2 at kbase + (lane>>4)*2.
    const int n     = lane & 15;    // column within tile
    const int khalf = lane >> 4;    // 0 -> k+{0,1}, 1 -> k+{2,3}
    const int r     = wave * 16 + n;
    const bool isA  = (n == 0);     // lanes carrying A row 0

    int idx;
    if (r < TT)          idx = targets[(size_t)b * TT + r];
    else if (r < ROWS)   idx = noises[(size_t)b * NN + (r - TT)];
    else                 idx = 0;   // padded row; contribution masked later

    const float* rowp = W + (size_t)idx * DD + khalf * 2;

    v8f c = {0.f, 0.f, 0.f, 0.f, 0.f, 0.f, 0.f, 0.f};

#pragma unroll 8
    for (int k = 0; k < DD; k += 4) {
        // B operand: per-lane float2 gather from this lane's W row
        v2f bv = *(const v2f*)(rowp + k);
        // A operand: broadcast LDS read on all lanes, branchless mask so only
        // lanes with n==0 (rows M=0 halves) carry the feature chunk.
        float a0 = fsh[k + khalf * 2 + 0];
        float a1 = fsh[k + khalf * 2 + 1];
        v2f av;
        av.x = isA ? a0 : 0.0f;
        av.y = isA ? a1 : 0.0f;
        // D = A x B + C  (FP32 WMMA, accumulate across K chunks)
        c = __builtin_amdgcn_wmma_f32_16x16x4_f32(
                /*neg_a=*/false, av, /*neg_b=*/false, bv,
                /*c_mod=*/(short)0, c, /*reuse_a=*/false, /*reuse_b=*/false);
    }

    // C layout: VGPR0 lanes 0-15 = C[0][n] = dot(W[rowIdx(n)], feature)
    float val = 0.0f;
    if (lane < 16) {
        float dot = c[0];
        if (r < TT)         val = log_sigmoid(dot);       // positive samples
        else if (r < ROWS)  val = log_sigmoid(-dot);      // negative samples
        // padded rows contribute 0
    }

    // Wave reduction (lanes >=16 hold 0)
#pragma unroll
    for (int off = 16; off > 0; off >>= 1)
        val += __shfl_xor(val, off, 32);

    if (lane == 0) wsum[wave] = val;
    __syncthreads();

    if (tid == 0) {
        float s = 0.0f;
#pragma unroll
        for (int w = 0; w < NWAVES; ++w) s += wsum[w];
        partial[b] = -s / (float)(TT + NN);
    }
}

__global__ __launch_bounds__(256)
void nsl_reduce_kernel(const float* __restrict__ partial, float* __restrict__ out) {
    __shared__ float sh[256];
    const int t = threadIdx.x;
    float s = 0.0f;
    for (int i = t; i < BB; i += 256) s += partial[i];
    sh[t] = s;
    __syncthreads();
    for (int stride = 128; stride > 0; stride >>= 1) {
        if (t < stride) sh[t] += sh[t + stride];
        __syncthreads();
    }
    if (t == 0) out[0] = sh[0];
}

extern "C" void kernel_launch(void* const* d_in, const int* in_sizes, int n_in,
                              void* d_out, int out_size, void* d_ws, size_t ws_size,
                              hipStream_t stream) {
    const float* feat    = (const float*)d_in[0];   // [B, D] f32
    const int*   targets = (const int*)d_in[1];     // [B, T] int
    const int*   noises  = (const int*)d_in[2];     // [B, N] int
    const float* W       = (const float*)d_in[3];   // [V, D] f32
    float* partial = (float*)d_ws;                  // 2048 floats of scratch
    float* out     = (float*)d_out;                 // scalar f32

    nsl_main_kernel<<<BB, BLOCK, 0, stream>>>(feat, targets, noises, W, partial);
    nsl_reduce_kernel<<<1, 256, 0, stream>>>(partial, out);
}